// PointHead_66013647339959
// MI455X (gfx1250) — compile-verified
//
#include <hip/hip_runtime.h>

#define B_    16
#define CO    21
#define CF    512
#define HW    16384
#define NOVER 12288
#define NCOV  1024
#define NIMP  3072
#define NPTS  4096
#define KDIM  533      // CO + CF
#define KPAD  544      // padded K used by the WMMA loop (multiple of 4)
#define KSTR  546      // LDS row stride (odd*2 -> conflict-free, 8B aligned)
#define MPAD  32       // padded M
#define SORTN 16384    // next pow2 >= NOVER
#define NT    64       // points per GEMM workgroup

typedef float v2f __attribute__((ext_vector_type(2)));
typedef float v8f __attribute__((ext_vector_type(8)));

typedef __attribute__((address_space(1))) int gint_t;  // global int
typedef __attribute__((address_space(3))) int lint_t;  // LDS int

__device__ __forceinline__ unsigned int ordf(float f) {
  unsigned int u = __float_as_uint(f);
  return (u & 0x80000000u) ? ~u : (u | 0x80000000u);
}

// ---------------------------------------------------------------------------
// Kernel 0: build zero-padded weight, M-major: Wt[MPAD][KPAD].
// Lane m reads Wt[m][k..k+1] as a single b64 in the GEMM loop.
// ---------------------------------------------------------------------------
__global__ void build_wt(const float* __restrict__ w, float* __restrict__ Wt) {
  int e = blockIdx.x * blockDim.x + threadIdx.x;
  if (e >= MPAD * KPAD) return;
  int m = e / KPAD;
  int k = e - m * KPAD;
  float v = 0.0f;
  if (m < CO && k < KDIM) v = w[m * KDIM + k];
  Wt[e] = v;
}

// ---------------------------------------------------------------------------
// Kernel 1: per-batch uncertainty + exact top-k via bitonic sort in LDS.
// One workgroup per batch. Dynamic LDS: SORTN * 8 bytes (128 KB).
// Key = (~ordered(uncertainty) << 32) | index  -> ascending sort gives
// descending uncertainty with stable lowest-index tie-break (top_k semantics).
// ---------------------------------------------------------------------------
__global__ void select_points(const float* __restrict__ outp,
                              const long long* __restrict__ over_idx,
                              const long long* __restrict__ cov_idx,
                              int* __restrict__ pts_ws,
                              float* __restrict__ pts_out) {
  extern __shared__ unsigned long long keys[];
  const int b = blockIdx.x;
  const int tid = threadIdx.x;
  const int nthr = blockDim.x;

  for (int i = tid; i < SORTN; i += nthr) {
    unsigned long long key = 0xFFFFFFFFFFFFFFFFull;  // padding sorts last
    if (i < NOVER) {
      int idx = (int)over_idx[(long long)b * NOVER + i];
      const float* base = outp + (long long)b * CO * HW + idx;
      float m1 = -3.4e38f, m2 = -3.4e38f;
#pragma unroll
      for (int c = 0; c < CO; ++c) {
        float v = base[c * HW];
        if (v > m1) { m2 = m1; m1 = v; }
        else if (v > m2) { m2 = v; }
      }
      float u = m2 - m1;                 // -(top1 - top2)
      unsigned int hi = ~ordf(u);        // ascending key == descending u
      key = ((unsigned long long)hi << 32) | (unsigned int)i;
    }
    keys[i] = key;
  }
  __syncthreads();

  for (int k = 2; k <= SORTN; k <<= 1) {
    for (int j = k >> 1; j > 0; j >>= 1) {
      for (int t = tid; t < (SORTN / 2); t += nthr) {
        int i = 2 * t - (t & (j - 1));
        int l = i + j;
        unsigned long long a = keys[i];
        unsigned long long c = keys[l];
        bool up = ((i & k) == 0);
        bool sw = up ? (a > c) : (a < c);
        if (sw) { keys[i] = c; keys[l] = a; }
      }
      __syncthreads();
    }
  }

  for (int pos = tid; pos < NPTS; pos += nthr) {
    int p;
    if (pos < NIMP) {
      int sel = (int)(keys[pos] & 0xFFFFFFFFull);
      p = (int)over_idx[(long long)b * NOVER + sel];
    } else {
      p = (int)cov_idx[(long long)b * NCOV + (pos - NIMP)];
    }
    pts_ws[b * NPTS + pos] = p;
    pts_out[b * NPTS + pos] = (float)p;   // tuple output #2, flattened as float
  }
}

// ---------------------------------------------------------------------------
// Kernel 2: async-gather feature tile into LDS, fp32 WMMA GEMM, bias, store.
// Grid: (NPTS/NT, B). Block: 256 threads (8 waves).
// All 8 waves issue GLOBAL_LOAD_ASYNC_TO_LDS_B32 gathers (ASYNCcnt);
// waves 0-3 then run dual-accumulator WMMA chains (M tiles 0-15 / 16-31).
// LDS F tile is point-major: fT[n][KSTR] so B fragments are one ds_load_b64.
// ---------------------------------------------------------------------------
__global__ void point_gemm(const float* __restrict__ outp,
                           const float* __restrict__ res2,
                           const float* __restrict__ Wt,
                           const float* __restrict__ bias,
                           const int* __restrict__ pts_ws,
                           float* __restrict__ rend) {
  extern __shared__ float smem[];
  float* fT  = smem;                      // [NT][KSTR]
  int*   pts = (int*)(smem + NT * KSTR);  // [NT]

  const int tile = blockIdx.x;
  const int b    = blockIdx.y;
  const int tid  = threadIdx.x;

  if (tid < NT) pts[tid] = pts_ws[b * NPTS + tile * NT + tid];
  __syncthreads();

  const float* cbase = outp + (long long)b * CO * HW;
  const float* fbase = res2 + (long long)b * CF * HW;

  // async scattered gather: rows 0..20 coarse, 21..532 fine
  for (int e = tid; e < KDIM * NT; e += blockDim.x) {
    int k = e >> 6;           // NT == 64: consecutive lanes share a plane
    int n = e & (NT - 1);
    const float* g = (k < CO) ? (cbase + k * HW + pts[n])
                              : (fbase + (k - CO) * HW + pts[n]);
    __builtin_amdgcn_global_load_async_to_lds_b32(
        (gint_t*)g, (lint_t*)&fT[n * KSTR + k], 0, 0);
  }
  // zero the K padding region (disjoint from async destinations)
  for (int e = tid; e < NT * (KSTR - KDIM); e += blockDim.x) {
    int n = e / (KSTR - KDIM);
    int k = KDIM + (e - n * (KSTR - KDIM));
    fT[n * KSTR + k] = 0.0f;
  }
  asm volatile("s_wait_asynccnt 0" ::: "memory");
  __syncthreads();

  const int wv = tid >> 5;
  if (wv < 4) {
    const int lane = tid & 31;
    const int half = lane >> 4;       // 0/1
    const int mr   = lane & 15;       // M row (A) / N col (B)
    const int coln = wv * 16 + mr;    // this wave's N subtile column

    const v2f* wa0 = (const v2f*)(Wt + mr * KPAD);         // M tile 0 row
    const v2f* wa1 = (const v2f*)(Wt + (16 + mr) * KPAD);  // M tile 1 row
    const v2f* fb  = (const v2f*)(fT + coln * KSTR);

    v8f c0 = {};                      // rows 0..15
    v8f c1 = {};                      // rows 16..31

    for (int k0 = 0; k0 < KPAD; k0 += 4) {
      int kv = (k0 >> 1) + half;      // v2f index at ka = k0 + half*2
      v2f a0 = wa0[kv];
      v2f a1 = wa1[kv];
      v2f bb = fb[kv];
      c0 = __builtin_amdgcn_wmma_f32_16x16x4_f32(false, a0, false, bb,
                                                 (short)0, c0, false, false);
      c1 = __builtin_amdgcn_wmma_f32_16x16x4_f32(false, a1, false, bb,
                                                 (short)0, c1, false, false);
    }

    // store D: lane l, reg r -> M = r + 8*(l>=16) (+16 for c1), N = l&15
    const int col = tile * NT + wv * 16 + (lane & 15);
    float* obase = rend + (long long)b * CO * NPTS + col;
#pragma unroll
    for (int r = 0; r < 8; ++r) {
      int m0 = r + half * 8;
      if (m0 < CO) obase[m0 * NPTS] = c0[r] + bias[m0];
      int m1 = 16 + r + half * 8;
      if (m1 < CO) obase[m1 * NPTS] = c1[r] + bias[m1];
    }
  }
}

// ---------------------------------------------------------------------------
extern "C" void kernel_launch(void* const* d_in, const int* in_sizes, int n_in,
                              void* d_out, int out_size, void* d_ws, size_t ws_size,
                              hipStream_t stream) {
  (void)in_sizes; (void)n_in; (void)out_size; (void)ws_size;
  // inputs: 0:x (unused) 1:res2 2:out 3:w 4:b 5:over_idx 6:cov_idx
  const float*     res2     = (const float*)d_in[1];
  const float*     outp     = (const float*)d_in[2];
  const float*     w        = (const float*)d_in[3];
  const float*     bias     = (const float*)d_in[4];
  const long long* over_idx = (const long long*)d_in[5];
  const long long* cov_idx  = (const long long*)d_in[6];

  float* rend    = (float*)d_out;                       // B*CO*NPTS floats
  float* pts_out = rend + (size_t)B_ * CO * NPTS;       // B*NPTS floats

  int*   pts_ws = (int*)d_ws;                           // B*NPTS ints
  float* Wt     = (float*)((char*)d_ws + (size_t)B_ * NPTS * sizeof(int));

  build_wt<<<(MPAD * KPAD + 255) / 256, 256, 0, stream>>>(w, Wt);

  select_points<<<B_, 512, SORTN * sizeof(unsigned long long), stream>>>(
      outp, over_idx, cov_idx, pts_ws, pts_out);

  size_t gemm_lds = (size_t)NT * KSTR * sizeof(float) + NT * sizeof(int);
  point_gemm<<<dim3(NPTS / NT, B_), 256, gemm_lds, stream>>>(
      outp, res2, Wt, bias, pts_ws, rend);
}